// WaveletTransformLayer_82849919140039
// MI455X (gfx1250) — compile-verified
//
#include <hip/hip_runtime.h>

typedef __attribute__((ext_vector_type(2))) float v2f;
typedef __attribute__((ext_vector_type(8))) float v8f;

#define NCOL 1116
#define X_ST 1172   // row span 8+1116+40=1164; 1172%64=20, gcd 4 -> conflict-free columns
#define D_ST 1204   // row span 1202;          1204%64=52, gcd 4 -> conflict-free columns
#define S_ST 612    // row span 8+561+40=609;  612%64=36,  gcd 4 -> conflict-free columns
#define NW   8      // waves per block (wave32)
#define BLK  256

__device__ __constant__ float c_DLO[8] = {
    -0.010597401784997278f, 0.032883011666982945f, 0.030841381835986965f,
    -0.18703481171888114f,  -0.02798376941698385f, 0.6308807679295904f,
    0.7148465705525415f,    0.23037781330885523f};

// dec_lo[i], zero outside [0,8)
__device__ __forceinline__ float selLO(int i) {
  return (i >= 0 && i < 8) ? c_DLO[i] : 0.0f;
}
// dec_hi[i] = (-1)^(i+1) * dec_lo[7-i], zero outside [0,8)
__device__ __forceinline__ float selHI(int i) {
  return (i >= 0 && i < 8) ? (((i & 1) ? 1.0f : -1.0f) * c_DLO[7 - i]) : 0.0f;
}

__device__ __forceinline__ v8f wmma_f32(v2f a, v2f b, v8f c) {
  // fp32 WMMA: D(16x16) = A(16x4) * B(4x16) + C
  return __builtin_amdgcn_wmma_f32_16x16x4_f32(false, a, false, b, (short)0, c,
                                               false, false);
}

// Zero buf[r][start .. start+width) for all 16 rows (guard-band maintenance).
__device__ __forceinline__ void zero_tail(float* buf, int stride, int start,
                                          int width, int tid) {
  for (int i = tid; i < 16 * width; i += BLK)
    buf[(i / width) * stride + start + (i % width)] = 0.0f;
}

// Forward DWT level: ca[r,j] = sum_i src[r,i]*dec_lo[2j+1-i]; cd with dec_hi.
// Reads i in [-6, inLen+38] -> covered by guard bands, no bounds checks.
__device__ __forceinline__ void dwt_level(
    const float* src, int sStr,
    float* dstA, int aStr, float* dstD, int dStr, int outLen,
    const v2f (&BL)[10], const v2f (&BH)[10], int lane, int wave) {
  const int m   = lane & 15;
  const int kb  = (lane >> 4) << 1;   // K base {0,2} per half-wave (A/B layout)
  const int hi8 = (lane >> 4) << 3;   // C/D row offset per half-wave
  const float* rp = src + m * sStr;
  const int nt = (outLen + 15) >> 4;
  for (int t = wave; t < nt; t += NW) {
    const int j0 = t << 4;
    v8f ca = {}, cd = {};
#pragma unroll
    for (int q = 0; q < 10; ++q) {
      const int i0 = 2 * j0 - 6 + 4 * q + kb;          // even -> 8B aligned
      const v2f A = *reinterpret_cast<const v2f*>(rp + i0);  // ds_load_b64
      ca = wmma_f32(A, BL[q], ca);
      cd = wmma_f32(A, BH[q], cd);
    }
    const int col = j0 + m;
    if (col < outLen) {
#pragma unroll
      for (int v = 0; v < 8; ++v) {
        dstA[(v + hi8) * aStr + col] = ca[v];
        dstD[(v + hi8) * dStr + col] = cd[v];
      }
    }
  }
}

// Inverse DWT level: y[r,j] = sum_t ca[r,t]*dec_lo[2t+1-j] + cd[r,t]*dec_hi[2t+1-j]
// Reads t in [0, M+7] -> covered by zeroed tails (which also implement the crop).
__device__ __forceinline__ void idwt_level(
    const float* srcA, int aStr, const float* srcD, int dStr,
    float* dst, int oStr, int outLen,
    const v2f (&BL)[3], const v2f (&BH)[3], int lane, int wave) {
  const int m   = lane & 15;
  const int kb  = (lane >> 4) << 1;
  const int hi8 = (lane >> 4) << 3;
  const float* ra = srcA + m * aStr;
  const float* rd = srcD + m * dStr;
  const int nt = (outLen + 15) >> 4;
  for (int t = wave; t < nt; t += NW) {
    const int j0 = t << 4;
    const int tb = (j0 >> 1) + kb;                     // even; 8B aligned in S, only 4B in Dd when dOff is odd
    v8f acc = {};
#pragma unroll
    for (int q = 0; q < 3; ++q) {
      const v2f Aa = *reinterpret_cast<const v2f*>(ra + tb + 4 * q);
      const v2f Ad = *reinterpret_cast<const v2f*>(rd + tb + 4 * q);
      acc = wmma_f32(Aa, BL[q], acc);
      acc = wmma_f32(Ad, BH[q], acc);
    }
    const int col = j0 + m;
    if (col < outLen) {
#pragma unroll
      for (int v = 0; v < 8; ++v) dst[(v + hi8) * oStr + col] = acc[v];
    }
  }
}

extern "C" __global__ __launch_bounds__(BLK) void wavelet7_wmma(
    const float* __restrict__ x, const float* __restrict__ rawt,
    float* __restrict__ out) {
  extern __shared__ float lds[];
  float* Xb  = lds;                  // 16 x X_ST
  float* Dd  = Xb + 16 * X_ST;       // 16 x D_ST (packed details + 8-float gaps)
  float* S0b = Dd + 16 * D_ST;       // 16 x S_ST
  float* S1b = S0b + 16 * S_ST;      // 16 x S_ST
  float* X  = Xb + 8;                // data pointers (8-float front guards)
  float* S0 = S0b + 8;
  float* S1 = S1b + 8;

  const int tid  = threadIdx.x;
  const int lane = tid & 31;
  const int wave = tid >> 5;

  // Per-lane Toeplitz filter fragments (B layout: lane&15 = N, half-wave = K rows)
  v2f BfL[10], BfH[10], BiL[3], BiH[3];
  {
    const int n  = lane & 15;
    const int kb = (lane >> 4) << 1;
#pragma unroll
    for (int q = 0; q < 10; ++q) {
      const int b = 2 * n + 7 - 4 * q - kb;  // h[2n+7-4q-k]
      BfL[q].x = selLO(b);     BfL[q].y = selLO(b - 1);
      BfH[q].x = selHI(b);     BfH[q].y = selHI(b - 1);
    }
#pragma unroll
    for (int q = 0; q < 3; ++q) {
      const int b = 8 * q + 2 * kb + 1 - n;  // h[8q+2k+1-n]
      BiL[q].x = selLO(b);     BiL[q].y = selLO(b + 2);
      BiH[q].x = selHI(b);     BiH[q].y = selHI(b + 2);
    }
  }

  const long long row0 = (long long)blockIdx.x * 16;
  const float* gsrc = x + row0 * NCOL;
  float* gdst = out + row0 * NCOL;

  const int fOut[7]    = {561, 284, 145, 76, 41, 24, 15};
  const int dOff[7]    = {0, 569, 861, 1014, 1098, 1147, 1179};  // +8 gaps
  const int dGapSt[7]  = {561, 853, 1006, 1090, 1139, 1171, 1194};

  // ---- stage 16x1116 tile: global -> LDS (float4) + zero all guard bands ----
  for (int i = tid; i < 16 * (NCOL / 4); i += BLK) {
    const int r = i / (NCOL / 4);
    const int c = i % (NCOL / 4);
    const float4 v = reinterpret_cast<const float4*>(gsrc + r * NCOL)[c];
    reinterpret_cast<float4*>(X + r * X_ST)[c] = v;
  }
  for (int i = tid; i < 16 * 48; i += BLK) {          // X: front 8 + tail 40
    const int r = i / 48, c = i % 48;
    Xb[r * X_ST + (c < 8 ? c : (1124 - 8 + c))] = 0.0f;
  }
  for (int i = tid; i < 16 * 8; i += BLK) {           // S0/S1 front guards
    const int r = i / 8, c = i % 8;
    S0b[r * S_ST + c] = 0.0f;
    S1b[r * S_ST + c] = 0.0f;
  }
  for (int i = tid; i < 16 * 56; i += BLK) {          // detail gaps (7 x 8)
    const int r = i / 56, g = (i % 56) >> 3, o = i & 7;
    Dd[r * D_ST + dGapSt[g] + o] = 0.0f;
  }
  __syncthreads();

  // ---- forward: 7 levels; approx ping-pongs S0/S1, details into Dd ----
  {
    const float* src = X;
    int sStr = X_ST;
    for (int l = 0; l < 7; ++l) {
      float* ad = (l & 1) ? S1 : S0;                  // a7 ends in S0
      dwt_level(src, sStr, ad, S_ST, Dd + dOff[l], D_ST, fOut[l],
                BfL, BfH, lane, wave);
      zero_tail(ad, S_ST, fOut[l], 40, tid);          // disjoint from live writes
      __syncthreads();
      src = ad; sStr = S_ST;
    }
  }

  // ---- soft threshold on a7 and all details (gap zeros stay zero) ----
  const float thr = fmaxf(rawt[0], 0.01f);
  for (int r = 0; r < 16; ++r)
    for (int c = tid; c < 1202; c += BLK) {
      float* p = Dd + r * D_ST + c;
      const float vv = *p;
      *p = copysignf(fmaxf(fabsf(vv) - thr, 0.0f), vv);
    }
  for (int i = tid; i < 16 * 15; i += BLK) {
    float* p = S0 + (i / 15) * S_ST + (i % 15);
    const float vv = *p;
    *p = copysignf(fmaxf(fabsf(vv) - thr, 0.0f), vv);
  }
  __syncthreads();

  // ---- inverse: 7 levels; crop via zeroing source tail [M, M+16) ----
  {
    float* a = S0;
    for (int i = 0; i < 7; ++i) {
      const int dl   = 6 - i;
      const int M    = fOut[dl];       // cropped coefficient length
      const int outL = 2 * M - 6;      // 24,42,76,146,284,562,1116
      if (i > 0) {                     // i==0: a7 tail already zeroed by forward
        zero_tail(a, S_ST, M, 16, tid);
        __syncthreads();
      }
      float* dst; int oStr;
      if (i == 6) { dst = X;  oStr = X_ST; }
      else        { dst = (i & 1) ? S0 : S1; oStr = S_ST; }
      idwt_level(a, S_ST, Dd + dOff[dl], D_ST, dst, oStr, outL,
                 BiL, BiH, lane, wave);
      __syncthreads();
      a = dst;
    }
  }

  // ---- store 16x1116 tile: LDS -> global (float4) ----
  for (int i = tid; i < 16 * (NCOL / 4); i += BLK) {
    const int r = i / (NCOL / 4);
    const int c = i % (NCOL / 4);
    const float4 v = reinterpret_cast<const float4*>(X + r * X_ST)[c];
    reinterpret_cast<float4*>(gdst + r * NCOL)[c] = v;
  }
}

extern "C" void kernel_launch(void* const* d_in, const int* in_sizes, int n_in,
                              void* d_out, int out_size, void* d_ws, size_t ws_size,
                              hipStream_t stream) {
  (void)n_in; (void)out_size; (void)d_ws; (void)ws_size;
  const float* x    = (const float*)d_in[0];
  const float* rawt = (const float*)d_in[1];
  float* out = (float*)d_out;

  const int rows   = in_sizes[0] / NCOL;  // 32768
  const int ntiles = rows / 16;           // 2048 workgroups
  const size_t shmem =
      (size_t)16 * (X_ST + D_ST + 2 * S_ST) * sizeof(float);  // ~230.4 KB

  // Opt in to >64KB dynamic LDS (CDNA5 WGP allows up to 320KB per workgroup).
  hipFuncSetAttribute((const void*)wavelet7_wmma,
                      hipFuncAttributeMaxDynamicSharedMemorySize, (int)shmem);

  wavelet7_wmma<<<dim3(ntiles), dim3(BLK), shmem, stream>>>(x, rawt, out);
}